// CosineSimilarityLoss_28321014349854
// MI455X (gfx1250) — compile-verified
//
#include <hip/hip_runtime.h>
#include <math.h>

// CosineSimilarityLoss: gather rows of a/b by label, per-row cosine,
// threshold >= 0.1, masked mean. Output: single f32 scalar.
//
// Stage 1 (bandwidth-bound, ~64MB stream @ 23.3 TB/s ~ 3us): one wave32 per
//   row, coalesced b128 loads (lane = 16 contiguous floats per matrix),
//   3 f32 accumulators, wave shfl_xor reduction, per-block partial
//   (masked sum, count) -> workspace. Grid padded to a multiple of 64 so
//   padding slots are written as 0.0 every call.
// Stage 2: one wave reduces the partials with V_WMMA_F32_16X16X4_F32
//   against an all-ones B matrix (64 values per WMMA issue, unconditional
//   b64 loads, uniform control flow), then writes total / max(count, 1).

#define DVEC 512           // feature dim (fixed by reference)
#define ROWS_PER_BLOCK 8   // 8 waves * 32 lanes = 256 threads/block
#define EPSV 1e-8f
#define MIN_THRESH 0.1f

typedef __attribute__((ext_vector_type(2))) float v2f;
typedef __attribute__((ext_vector_type(8))) float v8f;

__global__ __launch_bounds__(256)
void cos_rows_kernel(const float* __restrict__ a,
                     const float* __restrict__ b,
                     const int*   __restrict__ labels,
                     float* __restrict__ psum,
                     float* __restrict__ pcnt,
                     int N)
{
    const int lane = threadIdx.x & 31;
    const int wave = threadIdx.x >> 5;
    const int row  = blockIdx.x * ROWS_PER_BLOCK + wave;

    float ip = 0.0f, n1 = 0.0f, n2 = 0.0f;

    if (row < N) {
        const long long lbl = (long long)labels[row];
        // lane covers 16 contiguous floats: 4x b128 loads per matrix
        const float4* ra = (const float4*)(a + (size_t)lbl * DVEC) + lane * 4;
        const float4* rb = (const float4*)(b + (size_t)lbl * DVEC) + lane * 4;
#pragma unroll
        for (int i = 0; i < 4; ++i) {
            float4 x = ra[i];
            float4 y = rb[i];
            ip = fmaf(x.x, y.x, ip); ip = fmaf(x.y, y.y, ip);
            ip = fmaf(x.z, y.z, ip); ip = fmaf(x.w, y.w, ip);
            n1 = fmaf(x.x, x.x, n1); n1 = fmaf(x.y, x.y, n1);
            n1 = fmaf(x.z, x.z, n1); n1 = fmaf(x.w, x.w, n1);
            n2 = fmaf(y.x, y.x, n2); n2 = fmaf(y.y, y.y, n2);
            n2 = fmaf(y.z, y.z, n2); n2 = fmaf(y.w, y.w, n2);
        }
    }

    // wave32 reduction of the three partials
#pragma unroll
    for (int off = 16; off > 0; off >>= 1) {
        ip += __shfl_xor(ip, off, 32);
        n1 += __shfl_xor(n1, off, 32);
        n2 += __shfl_xor(n2, off, 32);
    }

    __shared__ float s_sum[ROWS_PER_BLOCK];
    __shared__ float s_cnt[ROWS_PER_BLOCK];

    if (lane == 0) {
        float contrib = 0.0f, cnt = 0.0f;
        if (row < N) {
            float denom = fmaxf(sqrtf(n1) * sqrtf(n2), EPSV);
            float cosv  = ip / denom;
            if (cosv >= MIN_THRESH) { contrib = cosv; cnt = 1.0f; }
        }
        s_sum[wave] = contrib;
        s_cnt[wave] = cnt;
    }
    __syncthreads();

    // Every block (including grid padding) writes its slot each call:
    // padded/overhanging blocks write exact 0.0 partials.
    if (threadIdx.x == 0) {
        float ts = 0.0f, tc = 0.0f;
#pragma unroll
        for (int i = 0; i < ROWS_PER_BLOCK; ++i) { ts += s_sum[i]; tc += s_cnt[i]; }
        psum[blockIdx.x] = ts;
        pcnt[blockIdx.x] = tc;
    }
}

// Single-wave finalize: WMMA tree-sum of Ppad partials (sum & count).
// Ppad is a multiple of 64 -> no bounds checks, uniform control flow,
// EXEC stays all-ones as WMMA requires.
// A-matrix 16x4 f32 layout: lane L -> row M=L%16, VGPR j holds K = 2*(L/16)+j.
// B = all ones -> D[m][n] += sum_k A[m][k] : every column accumulates row sums.
__global__ __launch_bounds__(32)
void finalize_wmma_kernel(const float* __restrict__ psum,
                          const float* __restrict__ pcnt,
                          float* __restrict__ out,
                          int Ppad)
{
    const int lane = threadIdx.x;

    v8f accs, accc;
#pragma unroll
    for (int r = 0; r < 8; ++r) { accs[r] = 0.0f; accc[r] = 0.0f; }

    v2f ones;
    ones[0] = 1.0f; ones[1] = 1.0f;

    // lane's element offset inside a 64-wide tile (always even -> 8B aligned)
    const int base_off = (lane & 15) * 4 + (lane >> 4) * 2;
    const v2f* ps2 = (const v2f*)(psum + base_off);
    const v2f* pc2 = (const v2f*)(pcnt + base_off);

    for (int t = 0; t < (Ppad >> 6); ++t) {
        v2f sa = ps2[t * 32];   // t*64 floats = t*32 v2f
        v2f ca = pc2[t * 32];
        // D = A x ones + C  (8 args: neg_a, A, neg_b, B, c_mod, C, reuse_a, reuse_b)
        accs = __builtin_amdgcn_wmma_f32_16x16x4_f32(
            false, sa, false, ones, (short)0, accs, false, false);
        accc = __builtin_amdgcn_wmma_f32_16x16x4_f32(
            false, ca, false, ones, (short)0, accc, false, false);
    }

    // C/D layout: this lane's column holds rows M = r (+8 for lanes 16..31)
    float s = 0.0f, c = 0.0f;
#pragma unroll
    for (int r = 0; r < 8; ++r) { s += accs[r]; c += accc[r]; }
    // combine M=0..7 half with M=8..15 half (columns are identical)
    s += __shfl_xor(s, 16, 32);
    c += __shfl_xor(c, 16, 32);

    if (lane == 0) {
        out[0] = s / fmaxf(c, 1.0f);
    }
}

extern "C" void kernel_launch(void* const* d_in, const int* in_sizes, int n_in,
                              void* d_out, int out_size, void* d_ws, size_t ws_size,
                              hipStream_t stream)
{
    const float* a      = (const float*)d_in[0];
    const float* b      = (const float*)d_in[1];
    const int*   labels = (const int*)d_in[2];   // jax default: int32
    float* out = (float*)d_out;

    const int N = in_sizes[2];
    const int blocks = (N + ROWS_PER_BLOCK - 1) / ROWS_PER_BLOCK;
    const int blocks_pad = (blocks + 63) & ~63;   // pad grid: extra blocks write 0.0

    float* psum = (float*)d_ws;
    float* pcnt = psum + blocks_pad;

    cos_rows_kernel<<<blocks_pad, 256, 0, stream>>>(a, b, labels, psum, pcnt, N);
    finalize_wmma_kernel<<<1, 32, 0, stream>>>(psum, pcnt, out, blocks_pad);
}